// PBDCorrectedVelocityStrategy_87239375716836
// MI455X (gfx1250) — compile-verified
//
#include <hip/hip_runtime.h>
#include <stdint.h>

// MI455X / gfx1250 implementation.
// Dominant cost: 32 batched 1024x1024 fp32 Cholesky factorizations (~11.4 GFLOP)
// via V_WMMA_F32_16X16X4_F32 trailing updates (KB=32 panel, 2x2 tile blocking,
// 32 WMMAs per inner iteration). The 128 MB of A matrices is L2-resident
// (192 MB L2). Panel tiles are streamed into LDS with the Tensor Data Mover
// (tensor_load_to_lds + s_wait_tensorcnt), one private 4 KB slot per wave.

typedef __attribute__((ext_vector_type(2))) float v2f;
typedef __attribute__((ext_vector_type(8))) float v8f;
typedef __attribute__((ext_vector_type(4))) unsigned int v4u;
typedef __attribute__((ext_vector_type(8))) int v8i;
typedef __attribute__((ext_vector_type(4))) int v4i;

#define BB 32
#define NN 512
#define EE 1024
#define DM 256
#define NB32 32 // EE / 32 superblocks

// ---------------------------------------------------------------------------
// TDM: DMA a 32x32 fp32 tile (row stride EE floats) from global into LDS.
// D# built per CDNA5 ISA (2D tile, groups 2/3 = 0).
// ---------------------------------------------------------------------------
__device__ __forceinline__ void tdm_load_tile_32x32(uint32_t lds_off, const float* gaddr)
{
#if __has_builtin(__builtin_amdgcn_tensor_load_to_lds)
    const uint64_t ga = (uint64_t)(uintptr_t)gaddr;
    v4u g0;
    g0[0] = 1u;                                   // count=1, user descriptor
    g0[1] = lds_off;                              // lds_addr (bytes)
    g0[2] = (uint32_t)(ga & 0xFFFFFFFFu);         // global_addr[31:0]
    g0[3] = (uint32_t)((ga >> 32) & 0x01FFFFFFu)  // global_addr[56:32]
            | (2u << 30);                         // type = 2 ("image")
    v8i g1;
    g1[0] = (int)(2u << 16);                      // data_size=4B; no mask/pad/iter
    g1[1] = (int)((uint32_t)(EE & 0xFFFF) << 16); // tensor_dim0[15:0] @ bits63:48
    g1[2] = (int)((uint32_t)(EE >> 16) | ((uint32_t)(EE & 0xFFFF) << 16)); // dim0 hi | dim1 lo
    g1[3] = (int)((uint32_t)(EE >> 16) | (32u << 16)); // dim1 hi | tile_dim0=32
    g1[4] = (int)32;                              // tile_dim1=32, tile_dim2=0
    g1[5] = (int)EE;                              // tensor_dim0_stride[31:0]
    g1[6] = 0;                                    // stride0 hi | stride1 lo
    g1[7] = 0;
    const v4i z4 = {0, 0, 0, 0};
#if defined(__clang_major__) && __clang_major__ >= 23
    const v8i z8 = {0, 0, 0, 0, 0, 0, 0, 0};
    __builtin_amdgcn_tensor_load_to_lds(g0, g1, z4, z4, z8, 0);
#else
    __builtin_amdgcn_tensor_load_to_lds(g0, g1, z4, z4, 0);
#endif
#endif
}

// ---------------------------------------------------------------------------
// Kernel 1: fused head GEMM + x_raw + 4 XPBD iterations + v, dx, rhs.
// One workgroup per batch; positions/velocities live in LDS.
// ---------------------------------------------------------------------------
__global__ __launch_bounds__(256) void k_head_xpbd(
    const float* __restrict__ kp_g, const float* __restrict__ ts,
    const float* __restrict__ h, const float* __restrict__ W,
    const float* __restrict__ bh, const int* __restrict__ eidx,
    const float* __restrict__ rl,
    float* __restrict__ v_ws, float* __restrict__ dx_ws, float* __restrict__ rhs_ws)
{
    const int b = blockIdx.x;
    const int tid = threadIdx.x;
    __shared__ float Wl[DM * 3];
    __shared__ float bl[3];
    __shared__ float xs[NN * 3];
    __shared__ float kps[NN * 3];
    __shared__ float dacc[NN * 3];
    __shared__ float vs[NN * 3];

    for (int k = tid; k < DM * 3; k += 256) Wl[k] = W[k];
    if (tid < 3) bl[tid] = bh[tid];
    for (int k = tid; k < NN * 3; k += 256) kps[k] = kp_g[(size_t)b * NN * 3 + k];
    __syncthreads();

    const float tau = fmaxf(1.0f - ts[b], 0.001f);

    // v_pred = h @ W + b ; x_raw = kp + tau*v_pred   (memory-bound, float4 loads)
    for (int n = tid; n < NN; n += 256) {
        const float4* hp = (const float4*)(h + ((size_t)b * NN + n) * DM);
        float a0 = 0.f, a1 = 0.f, a2 = 0.f;
        for (int k4 = 0; k4 < DM / 4; ++k4) {
            const float4 hv = hp[k4];
            const int k = k4 * 4;
            a0 += hv.x * Wl[(k + 0) * 3 + 0]; a1 += hv.x * Wl[(k + 0) * 3 + 1]; a2 += hv.x * Wl[(k + 0) * 3 + 2];
            a0 += hv.y * Wl[(k + 1) * 3 + 0]; a1 += hv.y * Wl[(k + 1) * 3 + 1]; a2 += hv.y * Wl[(k + 1) * 3 + 2];
            a0 += hv.z * Wl[(k + 2) * 3 + 0]; a1 += hv.z * Wl[(k + 2) * 3 + 1]; a2 += hv.z * Wl[(k + 2) * 3 + 2];
            a0 += hv.w * Wl[(k + 3) * 3 + 0]; a1 += hv.w * Wl[(k + 3) * 3 + 1]; a2 += hv.w * Wl[(k + 3) * 3 + 2];
        }
        xs[n * 3 + 0] = kps[n * 3 + 0] + tau * (a0 + bl[0]);
        xs[n * 3 + 1] = kps[n * 3 + 1] + tau * (a1 + bl[1]);
        xs[n * 3 + 2] = kps[n * 3 + 2] + tau * (a2 + bl[2]);
    }
    __syncthreads();

    // XPBD projection: 4 Jacobi iterations with LDS scatter
    for (int it = 0; it < 4; ++it) {
        for (int k = tid; k < NN * 3; k += 256) dacc[k] = 0.0f;
        __syncthreads();
        for (int e = tid; e < EE; e += 256) {
            const int i = eidx[e], j = eidx[EE + e];
            const float d0 = xs[i * 3 + 0] - xs[j * 3 + 0];
            const float d1 = xs[i * 3 + 1] - xs[j * 3 + 1];
            const float d2 = xs[i * 3 + 2] - xs[j * 3 + 2];
            const float dist = sqrtf(d0 * d0 + d1 * d1 + d2 * d2 + 1e-9f);
            const float C = dist - rl[e];
            const float invd = 1.0f / (dist + 1e-9f);
            const float lam = -C / (2.0f + 1e-9f);
            const float c0 = fminf(fmaxf(lam * d0 * invd, -0.15f), 0.15f);
            const float c1 = fminf(fmaxf(lam * d1 * invd, -0.15f), 0.15f);
            const float c2 = fminf(fmaxf(lam * d2 * invd, -0.15f), 0.15f);
            atomicAdd(&dacc[i * 3 + 0], -c0);
            atomicAdd(&dacc[i * 3 + 1], -c1);
            atomicAdd(&dacc[i * 3 + 2], -c2);
            atomicAdd(&dacc[j * 3 + 0], c0);
            atomicAdd(&dacc[j * 3 + 1], c1);
            atomicAdd(&dacc[j * 3 + 2], c2);
        }
        __syncthreads();
        for (int k = tid; k < NN * 3; k += 256) xs[k] += dacc[k];
        __syncthreads();
    }

    // v = (x_corr - kp)/tau
    const float itau = 1.0f / tau;
    for (int k = tid; k < NN * 3; k += 256) {
        const float vv = (xs[k] - kps[k]) * itau;
        vs[k] = vv;
        v_ws[(size_t)b * NN * 3 + k] = vv;
    }
    __syncthreads();

    // dx from KEYPOINTS (reference tangent-projects around keypoints), rhs = 2*dx.dv
    for (int e = tid; e < EE; e += 256) {
        const int i = eidx[e], j = eidx[EE + e];
        const float d0 = kps[i * 3 + 0] - kps[j * 3 + 0];
        const float d1 = kps[i * 3 + 1] - kps[j * 3 + 1];
        const float d2 = kps[i * 3 + 2] - kps[j * 3 + 2];
        const float e0 = vs[i * 3 + 0] - vs[j * 3 + 0];
        const float e1 = vs[i * 3 + 1] - vs[j * 3 + 1];
        const float e2 = vs[i * 3 + 2] - vs[j * 3 + 2];
        dx_ws[((size_t)b * EE + e) * 3 + 0] = d0;
        dx_ws[((size_t)b * EE + e) * 3 + 1] = d1;
        dx_ws[((size_t)b * EE + e) * 3 + 2] = d2;
        rhs_ws[(size_t)b * EE + e] = 2.0f * (d0 * e0 + d1 * e1 + d2 * e2);
    }
}

// ---------------------------------------------------------------------------
// Kernel 2: A = 4*(Dx Dx^T) .* sign + eps*I.
// Each wave computes one 16x16 tile: Gram part via one V_WMMA_F32_16X16X4_F32
// (K=3 padded to 4 with zeros), then elementwise sign/eps and store.
// A-op: A[m][k], m=lane%16, k=2*(lane/16)+comp; B symmetric; C/D: m=reg+8*hi,
// n=lane%16 (CDNA5 ISA §7.12.2).
// ---------------------------------------------------------------------------
__global__ __launch_bounds__(256) void k_build_A(
    const float* __restrict__ dx_ws, const int* __restrict__ eidx, float* __restrict__ A)
{
    const int b = blockIdx.z;
    const int wave = threadIdx.x >> 5;
    const int lane = threadIdx.x & 31;
    const int hi = lane >> 4, lo = lane & 15;
    const int e0 = blockIdx.x * 16;
    const int f0 = (blockIdx.y * 8 + wave) * 16;
    const float* dxb = dx_ws + (size_t)b * EE * 3;

    const int ka = 2 * hi; // k = ka, ka+1 ; k==3 is the zero pad
    v2f av, bv;
    av.x = dxb[(e0 + lo) * 3 + ka];
    av.y = (ka + 1 < 3) ? dxb[(e0 + lo) * 3 + ka + 1] : 0.0f;
    bv.x = dxb[(f0 + lo) * 3 + ka];
    bv.y = (ka + 1 < 3) ? dxb[(f0 + lo) * 3 + ka + 1] : 0.0f;
    v8f c = {};
    c = __builtin_amdgcn_wmma_f32_16x16x4_f32(false, av, false, bv, (short)0, c, false, false);

    const int f = f0 + lo;
    const int fi = eidx[f], fj = eidx[EE + f];
    float* Ab = A + (size_t)b * EE * EE;
#pragma unroll
    for (int r = 0; r < 8; ++r) {
        const int e = e0 + r + 8 * hi;
        const int ei = eidx[e], ej = eidx[EE + e];
        const float s = (float)((ei == fi) - (ei == fj) - (ej == fi) + (ej == fj));
        Ab[(size_t)e * EE + f] = 4.0f * c[r] * s + ((e == f) ? 1e-6f : 0.0f);
    }
}

// ---------------------------------------------------------------------------
// Kernel 3a: factor 32x32 diagonal block (column-parallel) + triangular-solve
// the panel rows below it. One workgroup per batch.
// ---------------------------------------------------------------------------
__global__ __launch_bounds__(256) void k_chol_panel(float* __restrict__ A, int kb)
{
    const int b = blockIdx.x;
    const int tid = threadIdx.x;
    float* M = A + (size_t)b * EE * EE;
    const int k0 = kb * 32;
    __shared__ float Lkk[32][33];
    for (int idx = tid; idx < 32 * 32; idx += 256)
        Lkk[idx >> 5][idx & 31] = M[(size_t)(k0 + (idx >> 5)) * EE + k0 + (idx & 31)];
    __syncthreads();
    // column-parallel 32x32 Cholesky
    for (int c = 0; c < 32; ++c) {
        if (tid == 0) {
            float s = Lkk[c][c];
            for (int u = 0; u < c; ++u) s -= Lkk[c][u] * Lkk[c][u];
            Lkk[c][c] = sqrtf(fmaxf(s, 1e-20f));
        }
        __syncthreads();
        const float inv = 1.0f / Lkk[c][c];
        const int r = c + 1 + tid;
        if (r < 32) {
            float t = Lkk[r][c];
            for (int u = 0; u < c; ++u) t -= Lkk[r][u] * Lkk[c][u];
            Lkk[r][c] = t * inv;
        }
        __syncthreads();
    }
    for (int idx = tid; idx < 32 * 32; idx += 256) {
        const int r = idx >> 5, cc = idx & 31;
        M[(size_t)(k0 + r) * EE + k0 + cc] = (cc <= r) ? Lkk[r][cc] : 0.0f;
    }
    // panel: L[row, kblk] = A[row, kblk] * inv(Lkk)^T, one row per thread
    const int nrows = EE - (k0 + 32);
    for (int rr = tid; rr < nrows; rr += 256) {
        const int row = k0 + 32 + rr;
        float* Mp = &M[(size_t)row * EE + k0];
        float a[32];
        for (int cc = 0; cc < 32; ++cc) a[cc] = Mp[cc];
        for (int cc = 0; cc < 32; ++cc) {
            float s = a[cc];
            for (int u = 0; u < cc; ++u) s -= a[u] * Lkk[cc][u];
            a[cc] = s / Lkk[cc][cc];
        }
        for (int cc = 0; cc < 32; ++cc) Mp[cc] = a[cc];
    }
}

// ---------------------------------------------------------------------------
// Kernel 3b: trailing SYRK A[ib,jb] -= P[ib] P[jb]^T, 32x32 superblocks.
// grid = (NB32-1-kb, B); blockIdx.x -> jb; 8 waves stride over ib.
// Each wave: TDM-loads its P[ib] 32x32 tile into a private 4 KB LDS slot
// (s_wait_tensorcnt 0, no cross-wave barrier needed), then 2x2 register-blocked
// WMMA: 32 x V_WMMA_F32_16X16X4_F32 per iteration. fp32 WMMA has no A/B negate
// (NEG=CNeg only), so the A operand is negated in registers.
// ---------------------------------------------------------------------------
__global__ __launch_bounds__(256) void k_chol_update(float* __restrict__ A, int kb)
{
    __shared__ float lds_panel[8 * 1024]; // 8 waves x (32x32 fp32) = 32 KB
    const int b = blockIdx.y;
    const int jb = kb + 1 + blockIdx.x;
    const int wave = threadIdx.x >> 5;
    const int lane = threadIdx.x & 31;
    const int hi = lane >> 4, lo = lane & 15;
    const int k0 = kb * 32;
    float* M = A + (size_t)b * EE * EE;

    // B operands (reused across the whole ib loop): B[k][n] = P[jb-rows][k]
    v2f bop[2][8];
#pragma unroll
    for (int jt = 0; jt < 2; ++jt) {
        const float* Pj = &M[(size_t)(jb * 32 + 16 * jt + lo) * EE + k0];
#pragma unroll
        for (int t = 0; t < 8; ++t) {
            bop[jt][t].x = Pj[4 * t + 2 * hi];
            bop[jt][t].y = Pj[4 * t + 2 * hi + 1];
        }
    }

    float* slotw = &lds_panel[wave * 1024];
    const uint32_t lds_off = (uint32_t)(uintptr_t)slotw;

    for (int ib = jb + wave; ib < NB32; ib += 8) {
        const float* gsrc = &M[(size_t)(ib * 32) * EE + k0];
#if __has_builtin(__builtin_amdgcn_tensor_load_to_lds)
        tdm_load_tile_32x32(lds_off, gsrc);
#if __has_builtin(__builtin_amdgcn_s_wait_tensorcnt)
        __builtin_amdgcn_s_wait_tensorcnt(0);
#endif
#else
        { // fallback: per-wave cooperative copy, row per lane
            const float4* s4 = (const float4*)(gsrc + (size_t)lane * EE);
            float4* d4 = (float4*)(slotw + lane * 32);
            for (int q = 0; q < 8; ++q) d4[q] = s4[q];
        }
#endif
        // negated A operands for both 16-row tiles of the superblock
        v2f aop[2][8];
#pragma unroll
        for (int it = 0; it < 2; ++it)
#pragma unroll
            for (int t = 0; t < 8; ++t) {
                const int row = 16 * it + lo;
                aop[it][t].x = -slotw[row * 32 + 4 * t + 2 * hi];
                aop[it][t].y = -slotw[row * 32 + 4 * t + 2 * hi + 1];
            }
#pragma unroll
        for (int it = 0; it < 2; ++it)
#pragma unroll
            for (int jt = 0; jt < 2; ++jt) {
                float* Ct = &M[(size_t)(ib * 32 + 16 * it) * EE + jb * 32 + 16 * jt];
                v8f c;
#pragma unroll
                for (int r = 0; r < 8; ++r) c[r] = Ct[(size_t)(r + 8 * hi) * EE + lo];
#pragma unroll
                for (int t = 0; t < 8; ++t)
                    c = __builtin_amdgcn_wmma_f32_16x16x4_f32(false, aop[it][t], false,
                                                              bop[jt][t], (short)0, c,
                                                              false, false);
#pragma unroll
                for (int r = 0; r < 8; ++r) Ct[(size_t)(r + 8 * hi) * EE + lo] = c[r];
            }
    }
}

// ---------------------------------------------------------------------------
// Kernel 4: cho_solve (forward + backward substitution, in-place in LDS) and
// final corr scatter into v. One workgroup per batch.
// ---------------------------------------------------------------------------
__global__ __launch_bounds__(256) void k_solve_scatter(
    const float* __restrict__ A, const float* __restrict__ rhs_ws,
    const float* __restrict__ dx_ws, const float* __restrict__ v_ws,
    const int* __restrict__ eidx, float* __restrict__ out)
{
    const int b = blockIdx.x;
    const int tid = threadIdx.x;
    const float* L = A + (size_t)b * EE * EE;
    __shared__ float s[EE];
    __shared__ float vsh[NN * 3];
    for (int k = tid; k < EE; k += 256) s[k] = rhs_ws[(size_t)b * EE + k];
    for (int k = tid; k < NN * 3; k += 256) vsh[k] = v_ws[(size_t)b * NN * 3 + k];
    __syncthreads();
    // forward: L y = rhs
    for (int c = 0; c < EE; ++c) {
        if (tid == 0) s[c] = s[c] / L[(size_t)c * EE + c];
        __syncthreads();
        const float yc = s[c];
        for (int r = c + 1 + tid; r < EE; r += 256)
            s[r] -= L[(size_t)r * EE + c] * yc;
        __syncthreads();
    }
    // backward: L^T x = y (coalesced row reads of L)
    for (int c = EE - 1; c >= 0; --c) {
        if (tid == 0) s[c] = s[c] / L[(size_t)c * EE + c];
        __syncthreads();
        const float xc = s[c];
        const float* Lrow = &L[(size_t)c * EE];
        for (int r = tid; r < c; r += 256)
            s[r] -= Lrow[r] * xc;
        __syncthreads();
    }
    // corr_e = 2*lam*dx ; v[i] -= corr_e, v[j] += corr_e
    for (int e = tid; e < EE; e += 256) {
        const int i = eidx[e], j = eidx[EE + e];
        const float l2 = 2.0f * s[e];
        const float c0 = l2 * dx_ws[((size_t)b * EE + e) * 3 + 0];
        const float c1 = l2 * dx_ws[((size_t)b * EE + e) * 3 + 1];
        const float c2 = l2 * dx_ws[((size_t)b * EE + e) * 3 + 2];
        atomicAdd(&vsh[i * 3 + 0], -c0);
        atomicAdd(&vsh[i * 3 + 1], -c1);
        atomicAdd(&vsh[i * 3 + 2], -c2);
        atomicAdd(&vsh[j * 3 + 0], c0);
        atomicAdd(&vsh[j * 3 + 1], c1);
        atomicAdd(&vsh[j * 3 + 2], c2);
    }
    __syncthreads();
    for (int k = tid; k < NN * 3; k += 256) out[(size_t)b * NN * 3 + k] = vsh[k];
}

// ---------------------------------------------------------------------------
extern "C" void kernel_launch(void* const* d_in, const int* in_sizes, int n_in,
                              void* d_out, int out_size, void* d_ws, size_t ws_size,
                              hipStream_t stream)
{
    const float* kp = (const float*)d_in[0];   // [B,N,3]
    const float* ts = (const float*)d_in[1];   // [B]
    const float* h  = (const float*)d_in[2];   // [B,N,256]
    const float* W  = (const float*)d_in[3];   // [256,3]
    const float* bh = (const float*)d_in[4];   // [3]
    const int*   ei = (const int*)d_in[5];     // [2,E] int32
    const float* rl = (const float*)d_in[6];   // [E]
    float* out = (float*)d_out;                // [B,N,3]

    // workspace layout (floats): v | dx | rhs | A   (~135 MB total)
    float* ws     = (float*)d_ws;
    float* v_ws   = ws;
    float* dx_ws  = v_ws + (size_t)BB * NN * 3;
    float* rhs_ws = dx_ws + (size_t)BB * EE * 3;
    float* Amat   = rhs_ws + (size_t)BB * EE;

    k_head_xpbd<<<dim3(BB), 256, 0, stream>>>(kp, ts, h, W, bh, ei, rl,
                                              v_ws, dx_ws, rhs_ws);
    k_build_A<<<dim3(EE / 16, (EE / 16) / 8, BB), 256, 0, stream>>>(dx_ws, ei, Amat);

    // right-looking blocked Cholesky (KB=32), host-driven so trailing updates
    // get grid-wide parallelism (up to 31x32 superblock-columns x batches)
    for (int kb = 0; kb < NB32; ++kb) {
        k_chol_panel<<<dim3(BB), 256, 0, stream>>>(Amat, kb);
        const int m = NB32 - 1 - kb;
        if (m > 0) k_chol_update<<<dim3(m, BB), 256, 0, stream>>>(Amat, kb);
    }

    k_solve_scatter<<<dim3(BB), 256, 0, stream>>>(Amat, rhs_ws, dx_ws, v_ws, ei, out);
}